// MeshPool_17566416241246
// MI455X (gfx1250) — compile-verified
//
#include <hip/hip_runtime.h>
#include <stdint.h>

// Problem constants (match setup_inputs)
#define BB 4
#define NN 8192
#define FF 64
#define EE 65536
#define KK 4096          // N/2 survivors
#define TGT 4096         // kills target
#define EPSB 0.05f

typedef float v2f __attribute__((ext_vector_type(2)));
typedef float v8f __attribute__((ext_vector_type(8)));

// ---------------------------------------------------------------- zero output
__global__ void k_zero(float* p, size_t n) {
    size_t i = (size_t)blockIdx.x * blockDim.x + threadIdx.x;
    size_t stride = (size_t)gridDim.x * blockDim.x;
    for (; i < n; i += stride) p[i] = 0.0f;
}

// ------------------------------------------------- vertex norms via f32 WMMA
// One wave32 per 16-vertex tile: D = A * A^T accumulated over K=4 chunks,
// diagonal of D = squared row norms. A(16x4,f32) and B(4x16)=A^T have the
// same lane/VGPR layout on CDNA5, so the same register pair feeds both.
__global__ void k_norms_wmma(const float* __restrict__ image,
                             float* __restrict__ norms) {
    int wave = (blockIdx.x * blockDim.x + threadIdx.x) >> 5;   // tile id
    int lane = threadIdx.x & 31;
    int row0 = wave * 16;                     // global row in [0, B*N)
    int m = lane & 15;
    int khalf = lane >> 4;                    // 0: K={0,1}, 1: K={2,3}
    const float* base = image + (size_t)(row0 + m) * FF;
    v8f c = {};
#pragma unroll
    for (int kc = 0; kc < FF / 4; ++kc) {
        int kb = kc * 4 + khalf * 2;
        v2f a;
        a.x = base[kb];
        a.y = base[kb + 1];
        c = __builtin_amdgcn_wmma_f32_16x16x4_f32(false, a, false, a,
                                                  (short)0, c, false, false);
    }
    // diagonal: D[r][r] -> lane r, vgpr r (r<8); D[8+r][8+r] -> lane 24+r, vgpr r
#pragma unroll
    for (int r = 0; r < 8; ++r) {
        if (lane == r)      norms[row0 + r]     = c[r];
        if (lane == 24 + r) norms[row0 + 8 + r] = c[r];
    }
}

// ------------------------------------------------------- edge keys + ok flag
__device__ __forceinline__ bool bnd_pt(const float* p) {
    return (p[0] < EPSB) | (p[0] > 1.0f - EPSB) | (p[1] < EPSB) | (p[1] > 1.0f - EPSB);
}

__global__ void k_edge_keys(const float* __restrict__ norms,
                            const float* __restrict__ vs,
                            const int* __restrict__ edges,
                            uint64_t* __restrict__ keys,
                            unsigned char* __restrict__ ok) {
    int tid = blockIdx.x * blockDim.x + threadIdx.x;      // [0, B*E)
    int b = tid >> 16;
    int e = tid & (EE - 1);
    const int* eb = edges + (size_t)b * 2 * EE;
    int v0 = eb[e];
    int v1 = eb[EE + e];
    float cost = norms[b * NN + v0] + norms[b * NN + v1];
    bool okf = !bnd_pt(vs + ((size_t)b * NN + v0) * 2) &&
               !bnd_pt(vs + ((size_t)b * NN + v1) * 2);
    ok[tid] = okf ? 1 : 0;
    unsigned u = __float_as_uint(cost);
    u = (u & 0x80000000u) ? ~u : (u | 0x80000000u);       // order-preserving
    keys[tid] = ((uint64_t)u << 32) | (unsigned)e;        // stable tie-break
}

// ---------------------------------------------------------- bitonic sort step
__global__ void k_bitonic(uint64_t* __restrict__ keys, unsigned j, unsigned k) {
    int tid = blockIdx.x * blockDim.x + threadIdx.x;      // [0, B*E)
    int b = tid >> 16;
    unsigned i = (unsigned)(tid & (EE - 1));
    unsigned l = i ^ j;
    if (l > i) {
        size_t o = (size_t)b << 16;
        uint64_t ki = keys[o + i];
        uint64_t kl = keys[o + l];
        bool asc = ((i & k) == 0);
        if (asc ? (ki > kl) : (ki < kl)) {
            keys[o + i] = kl;
            keys[o + l] = ki;
        }
    }
}

// -------------------------------------------------- sequential greedy collapse
// One wave32 per batch. Alive bitmask in LDS; each 32-edge chunk is loaded
// cooperatively, then replayed serially (decisions uniform across lanes).
__global__ void k_greedy(const uint64_t* __restrict__ keys,
                         const int* __restrict__ edges,
                         const unsigned char* __restrict__ ok,
                         int* __restrict__ mmap,
                         unsigned char* __restrict__ maskb) {
    __shared__ unsigned aliveS[NN / 32];
    __shared__ int sA[32], sB[32], sOk[32];
    int b = blockIdx.x;
    int lane = threadIdx.x;
    size_t bOff = (size_t)b * NN;
    const int* v0e = edges + (size_t)b * 2 * EE;
    const int* v1e = v0e + EE;
    const uint64_t* kb = keys + ((size_t)b << 16);
    const unsigned char* okb = ok + ((size_t)b << 16);

    for (int i = lane; i < NN / 32; i += 32) aliveS[i] = 0xFFFFFFFFu;
    for (int v = lane; v < NN; v += 32) mmap[bOff + v] = v;   // identity map
    __syncthreads();

    int nk = 0;
    for (int base = 0; base < EE; base += 32) {
        if (nk >= TGT) break;
        uint64_t key = kb[base + lane];
        int eid = (int)(key & 0xFFFFFFFFull);
        sA[lane] = v0e[eid];
        sB[lane] = v1e[eid];
        sOk[lane] = okb[eid];
        __syncthreads();
        for (int t = 0; t < 32; ++t) {
            if (nk >= TGT) break;
            if (!sOk[t]) continue;
            int a = sA[t];
            int d = sB[t];
            unsigned wa = aliveS[a >> 5];
            if (!((wa >> (a & 31)) & 1u)) continue;
            unsigned wd = aliveS[d >> 5];
            if (!((wd >> (d & 31)) & 1u)) continue;
            aliveS[d >> 5] = wd & ~(1u << (d & 31));   // uniform same-value write
            if (lane == 0) mmap[bOff + d] = a;
            ++nk;
        }
        __syncthreads();
    }
    for (int v = lane; v < NN; v += 32)
        maskb[bOff + v] = (unsigned char)((aliveS[v >> 5] >> (v & 31)) & 1u);
}

// ------------------------------------- survivor ranks, alive list, target map
__global__ void k_rank(const unsigned char* __restrict__ maskb,
                       const int* __restrict__ mmap,
                       int* __restrict__ tcol,
                       int* __restrict__ aliveList) {
    __shared__ int sums[1024];
    __shared__ int rankS[NN];
    int b = blockIdx.x;
    int t = threadIdx.x;
    size_t bOff = (size_t)b * NN;
    int base = t * 8;
    int mq[8];
    int c = 0;
#pragma unroll
    for (int q = 0; q < 8; ++q) { mq[q] = maskb[bOff + base + q]; c += mq[q]; }
    sums[t] = c;
    __syncthreads();
    for (int off = 1; off < 1024; off <<= 1) {     // Hillis-Steele inclusive
        int v = sums[t];
        int prev = (t >= off) ? sums[t - off] : 0;
        __syncthreads();
        sums[t] = v + prev;
        __syncthreads();
    }
    int r = sums[t] - c;                           // exclusive prefix
#pragma unroll
    for (int q = 0; q < 8; ++q) {
        int v = base + q;
        if (mq[q]) {
            rankS[v] = r;
            if (r < KK) aliveList[(size_t)b * KK + r] = v;
            ++r;
        } else {
            rankS[v] = 0x7fffffff;
        }
    }
    __syncthreads();
#pragma unroll
    for (int q = 0; q < 8; ++q) {
        int v = base + q;
        int par = mmap[bOff + v];
        int tv = -1;
        if (maskb[bOff + par]) {
            int rp = rankS[par];
            if (rp < KK) tv = rp;
        }
        tcol[bOff + v] = tv;
    }
}

// ----------------------------------------------------------- feature outputs
__global__ void k_feat_init(const float* __restrict__ image,
                            const int* __restrict__ aliveList,
                            float* __restrict__ feat) {
    int tid = blockIdx.x * blockDim.x + threadIdx.x;   // [0, B*K*F)
    int f = tid & (FF - 1);
    int p = (tid >> 6) & (KK - 1);
    int b = tid >> 18;
    int v = aliveList[(size_t)b * KK + p];
    feat[tid] = image[((size_t)b * NN + v) * FF + f];
}

__global__ void k_feat_scatter(const float* __restrict__ image,
                               const int* __restrict__ mmap,
                               const int* __restrict__ tcol,
                               float* __restrict__ feat) {
    int tid = blockIdx.x * blockDim.x + threadIdx.x;   // [0, B*N*F)
    int f = tid & (FF - 1);
    int v = (tid >> 6) & (NN - 1);
    int b = tid >> 19;
    size_t bOff = (size_t)b * NN;
    int par = mmap[bOff + v];
    if (par == v) return;                              // alive: nothing to add
    int tv = tcol[bOff + v];                           // rank of absorber (if selected)
    if (tv < 0) return;
    atomicAdd(&feat[((size_t)b * KK + tv) * FF + f], image[tid]);
}

// -------------------------------------------------------- adjacency pooling
// A2[t(i)][t(j)] |= A[i][j]; output gathered rows/cols, diagonal cleared.
__global__ void k_adj(const unsigned char* __restrict__ adjIn,
                      const int* __restrict__ tcol,
                      float* __restrict__ adjOut) {
    int tid = blockIdx.x * blockDim.x + threadIdx.x;   // [0, B*N*512)
    int c = tid & 511;                                 // 16-byte chunk in row
    int i = (tid >> 9) & (NN - 1);
    int b = tid >> 22;
    int ti = tcol[(size_t)b * NN + i];
    if (ti < 0) return;
    const uint4* rp = reinterpret_cast<const uint4*>(
        adjIn + ((size_t)b * NN + i) * NN);
    uint4 w = rp[c];
    unsigned arr[4] = {w.x, w.y, w.z, w.w};
    float* orow = adjOut + ((size_t)b * KK + ti) * KK;
    const int* tc = tcol + (size_t)b * NN;
    int jbase = c * 16;
#pragma unroll
    for (int d = 0; d < 4; ++d) {
        unsigned x = arr[d];
        if (!x) continue;
#pragma unroll
        for (int by = 0; by < 4; ++by) {
            if ((x >> (by * 8)) & 0xFFu) {
                int j = jbase + d * 4 + by;
                int tj = tc[j];
                if (tj >= 0 && tj != ti) orow[tj] = 1.0f;  // same-value races OK
            }
        }
    }
}

// ------------------------------------------------------------------ launcher
extern "C" void kernel_launch(void* const* d_in, const int* in_sizes, int n_in,
                              void* d_out, int out_size, void* d_ws, size_t ws_size,
                              hipStream_t stream) {
    const unsigned char* adj = (const unsigned char*)d_in[0];  // [B,N,N] bool
    const float* image = (const float*)d_in[1];                // [B,N,F]
    const float* vs = (const float*)d_in[2];                   // [B,N,2]
    const int* edges = (const int*)d_in[3];                    // [B,2,E]

    char* ws = (char*)d_ws;
    float* norms          = (float*)(ws + 0x000000);   // B*N*4      = 128KB
    uint64_t* keys        = (uint64_t*)(ws + 0x020000);// B*E*8      = 2MB
    unsigned char* ok     = (unsigned char*)(ws + 0x220000); // B*E  = 256KB
    int* mmap             = (int*)(ws + 0x260000);     // B*N*4      = 128KB
    unsigned char* maskb  = (unsigned char*)(ws + 0x280000); // B*N  = 32KB
    int* tcol             = (int*)(ws + 0x288000);     // B*N*4      = 128KB
    int* aliveList        = (int*)(ws + 0x2A8000);     // B*K*4      = 64KB

    float* adjOut = (float*)d_out;                       // B*K*K floats
    float* featOut = adjOut + (size_t)BB * KK * KK;      // B*K*F floats

    // 1. zero pooled adjacency
    k_zero<<<4096, 256, 0, stream>>>(adjOut, (size_t)BB * KK * KK);

    // 2. per-vertex squared norms via f32 WMMA (B*N/16 tiles, 8 waves/block)
    k_norms_wmma<<<(BB * NN / 16) / 8, 256, 0, stream>>>(image, norms);

    // 3. edge cost keys + interior flag
    k_edge_keys<<<(BB * EE) / 256, 256, 0, stream>>>(norms, vs, edges, keys, ok);

    // 4. stable ascending bitonic sort of (cost, edge-id) keys per batch
    for (unsigned k = 2; k <= EE; k <<= 1)
        for (unsigned j = k >> 1; j > 0; j >>= 1)
            k_bitonic<<<(BB * EE) / 256, 256, 0, stream>>>(keys, j, k);

    // 5. sequential greedy collapse, one wave per batch
    k_greedy<<<BB, 32, 0, stream>>>(keys, edges, ok, mmap, maskb);

    // 6. survivor ranks / alive list / per-vertex target map
    k_rank<<<BB, 1024, 0, stream>>>(maskb, mmap, tcol, aliveList);

    // 7. gather surviving features, then scatter-add collapsed ones
    k_feat_init<<<(BB * KK * FF) / 256, 256, 0, stream>>>(image, aliveList, featOut);
    k_feat_scatter<<<(BB * NN * FF) / 256, 256, 0, stream>>>(image, mmap, tcol, featOut);

    // 8. pooled adjacency
    k_adj<<<(BB * NN * (NN / 16)) / 256, 256, 0, stream>>>(adj, tcol, adjOut);
}